// EdgeAwareGCNEncoder_14431090115066
// MI455X (gfx1250) — compile-verified
//
#include <hip/hip_runtime.h>
#include <math.h>

#define NN 50000
#define NE 800000

typedef __attribute__((ext_vector_type(2))) float v2f;
typedef __attribute__((ext_vector_type(8))) float v8f;

__device__ __forceinline__ int f2ord(float f) {
    int i = __float_as_int(f);
    return (i >= 0) ? i : (i ^ 0x7fffffff);
}
__device__ __forceinline__ float ord2f(int i) {
    int j = (i >= 0) ? i : (i ^ 0x7fffffff);
    return __int_as_float(j);
}

// ---------------------------------------------------------------------------
// Y[Nrows x M] = A[Nrows x K] @ W[K x M] + bias[M]   (optional relu)
// One wave computes one 16x16 output tile with V_WMMA_F32_16X16X4_F32,
// stepping K by 4. fp32 in, fp32 accumulate — full precision.
// A 16x4 layout: lanes 0-15 hold K=0,1; lanes 16-31 hold K=2,3 (2 VGPRs).
// B 4x16 layout: mirrored (rows striped across lanes).
// C/D layout: vgpr r -> row r (lanes 0-15) / row r+8 (lanes 16-31).
// ---------------------------------------------------------------------------
__global__ void wmma_gemm_bias(const float* __restrict__ A,
                               const float* __restrict__ W,
                               const float* __restrict__ bias,
                               float* __restrict__ Y,
                               int Nrows, int K, int M, int do_relu) {
    int wid  = (blockIdx.x * blockDim.x + threadIdx.x) >> 5;
    int lane = threadIdx.x & 31;
    int tilesM = M >> 4;
    int tile_m = wid % tilesM;
    int tile_n = wid / tilesM;
    int row0 = tile_n << 4;
    if (row0 >= Nrows) return;
    int col0 = tile_m << 4;
    int half = lane >> 4;     // 0: K pair {0,1}, 1: K pair {2,3}
    int r    = lane & 15;

    v8f acc = {};
    const float* arow = A + (size_t)(row0 + r) * K + 2 * half;
    const float* wcol = W + (size_t)(2 * half) * M + col0 + r;
    for (int k = 0; k < K; k += 4) {
        v2f a = *(const v2f*)(arow + k);
        v2f b;
        b.x = wcol[(size_t)k * M];
        b.y = wcol[(size_t)(k + 1) * M];
        acc = __builtin_amdgcn_wmma_f32_16x16x4_f32(
            /*neg_a=*/false, a, /*neg_b=*/false, b,
            /*c_mod=*/(short)0, acc, /*reuse_a=*/false, /*reuse_b=*/false);
    }
    float bb = bias[col0 + r];
#pragma unroll
    for (int i = 0; i < 8; ++i) {
        float o = acc[i] + bb;
        if (do_relu) o = fmaxf(o, 0.0f);
        Y[(size_t)(row0 + i + 8 * half) * M + col0 + r] = o;
    }
}

// ---------------------------------------------------------------------------
// Edge pass 1: alpha[e,h] = q[dst] . (k[src] + edge_attr@We) / sqrt(64)
// plus ordered-int atomicMax per (dst, head). Wave-per-edge; each lane owns
// DTOT/32 contiguous channels; edge_attr broadcast via __shfl; We in LDS.
// ---------------------------------------------------------------------------
template <int H, int DTOT>
__global__ void edge_alpha(const int* __restrict__ src, const int* __restrict__ dst,
                           const float* __restrict__ ea, const float* __restrict__ We,
                           const float* __restrict__ q, const float* __restrict__ k,
                           float* __restrict__ alpha, int* __restrict__ mord, int E) {
    constexpr int CPL = DTOT / 32;
    constexpr int LPH = 32 / H;
    __shared__ float sWe[16 * DTOT];
    for (int i = threadIdx.x; i < 16 * DTOT; i += blockDim.x) sWe[i] = We[i];
    __syncthreads();

    int w    = (blockIdx.x * blockDim.x + threadIdx.x) >> 5;
    int lane = threadIdx.x & 31;
    if (w >= E) return;
    int s = src[w], d = dst[w];
    int chan = lane * CPL;

    float eav = (lane < 16) ? ea[(size_t)w * 16 + lane] : 0.0f;
    float ev[CPL];
#pragma unroll
    for (int c = 0; c < CPL; ++c) ev[c] = 0.0f;
#pragma unroll
    for (int j = 0; j < 16; ++j) {
        float ej = __shfl(eav, j, 32);
#pragma unroll
        for (int c = 0; c < CPL; ++c) ev[c] += ej * sWe[j * DTOT + chan + c];
    }
    const float* qd = q + (size_t)d * DTOT + chan;
    const float* ks = k + (size_t)s * DTOT + chan;
    float part = 0.0f;
#pragma unroll
    for (int c = 0; c < CPL; ++c) part += qd[c] * (ks[c] + ev[c]);
#pragma unroll
    for (int m = LPH >> 1; m >= 1; m >>= 1) part += __shfl_xor(part, m, 32);
    if ((lane & (LPH - 1)) == 0) {
        int head = lane / LPH;
        float al = part * 0.125f;  // 1/sqrt(64)
        alpha[(size_t)w * H + head] = al;
        atomicMax(&mord[d * H + head], f2ord(al));
    }
}

// Edge pass 3: scatter-add  a * (v[src] + e_vec)  into agg[dst].
template <int H, int DTOT>
__global__ void edge_agg(const int* __restrict__ src, const int* __restrict__ dst,
                         const float* __restrict__ ea, const float* __restrict__ We,
                         const float* __restrict__ v, const float* __restrict__ alpha,
                         const float* __restrict__ den, float* __restrict__ agg, int E) {
    constexpr int CPL = DTOT / 32;
    constexpr int LPH = 32 / H;
    __shared__ float sWe[16 * DTOT];
    for (int i = threadIdx.x; i < 16 * DTOT; i += blockDim.x) sWe[i] = We[i];
    __syncthreads();

    int w    = (blockIdx.x * blockDim.x + threadIdx.x) >> 5;
    int lane = threadIdx.x & 31;
    if (w >= E) return;
    int s = src[w], d = dst[w];
    int chan = lane * CPL;
    int head = lane / LPH;

    float eav = (lane < 16) ? ea[(size_t)w * 16 + lane] : 0.0f;
    float ev[CPL];
#pragma unroll
    for (int c = 0; c < CPL; ++c) ev[c] = 0.0f;
#pragma unroll
    for (int j = 0; j < 16; ++j) {
        float ej = __shfl(eav, j, 32);
#pragma unroll
        for (int c = 0; c < CPL; ++c) ev[c] += ej * sWe[j * DTOT + chan + c];
    }
    float a = alpha[(size_t)w * H + head] / (den[d * H + head] + 1e-16f);
    const float* vs = v + (size_t)s * DTOT + chan;
    float* ag = agg + (size_t)d * DTOT + chan;
#pragma unroll
    for (int c = 0; c < CPL; ++c) atomicAdd(&ag[c], a * (vs[c] + ev[c]));
}

// Edge pass 2: ex = exp(alpha - m[dst]); alpha <- ex; den[dst] += ex.
__global__ void edge_exp(const int* __restrict__ dst, float* __restrict__ alpha,
                         const float* __restrict__ mf, float* __restrict__ den,
                         int E, int H) {
    int idx = blockIdx.x * blockDim.x + threadIdx.x;
    if (idx >= E * H) return;
    int e = idx / H;
    int h = idx - e * H;
    int d = dst[e];
    float ex = __expf(alpha[idx] - mf[d * H + h]);
    alpha[idx] = ex;
    atomicAdd(&den[d * H + h], ex);
}

__global__ void init_m_den(int* __restrict__ mord, float* __restrict__ den, int n) {
    int i = blockIdx.x * blockDim.x + threadIdx.x;
    if (i < n) { mord[i] = 0x807fffff; /* f2ord(-inf) */ den[i] = 0.0f; }
}

// Convert ordered-int max back to float in place; empty segments (-inf) -> 0.
__global__ void finalize_m(int* __restrict__ mord, int n) {
    int i = blockIdx.x * blockDim.x + threadIdx.x;
    if (i < n) {
        float f = ord2f(mord[i]);
        if (!isfinite(f)) f = 0.0f;
        ((float*)mord)[i] = f;
    }
}

__global__ void relu_copy(const float* __restrict__ in, float* __restrict__ out, int n) {
    int i = blockIdx.x * blockDim.x + threadIdx.x;
    if (i < n) out[i] = fmaxf(in[i], 0.0f);
}

// ---------------------------------------------------------------------------
extern "C" void kernel_launch(void* const* d_in, const int* in_sizes, int n_in,
                              void* d_out, int out_size, void* d_ws, size_t ws_size,
                              hipStream_t stream) {
    const float* x     = (const float*)d_in[0];
    const int*   ei    = (const int*)d_in[1];
    const float* eattr = (const float*)d_in[2];
    const float *Wq1 = (const float*)d_in[3],  *bq1 = (const float*)d_in[4];
    const float *Wk1 = (const float*)d_in[5],  *bk1 = (const float*)d_in[6];
    const float *Wv1 = (const float*)d_in[7],  *bv1 = (const float*)d_in[8];
    const float *We1 = (const float*)d_in[9];
    const float *Ws1 = (const float*)d_in[10], *bs1 = (const float*)d_in[11];
    const float *Wq2 = (const float*)d_in[12], *bq2 = (const float*)d_in[13];
    const float *Wk2 = (const float*)d_in[14], *bk2 = (const float*)d_in[15];
    const float *Wv2 = (const float*)d_in[16], *bv2 = (const float*)d_in[17];
    const float *We2 = (const float*)d_in[18];
    const float *Ws2 = (const float*)d_in[19], *bs2 = (const float*)d_in[20];

    const int* src = ei;
    const int* dst = ei + NE;

    // workspace layout (all 4-byte elements)
    float* ws     = (float*)d_ws;
    float* qbuf   = ws;                         // NN*128
    float* kbuf   = qbuf + (size_t)NN * 128;    // NN*128
    float* vbuf   = kbuf + (size_t)NN * 128;    // NN*128
    float* hbuf   = vbuf + (size_t)NN * 128;    // NN*128 (layer-1 output)
    float* aggbuf = hbuf + (size_t)NN * 128;    // NN*128 (layer-1 agg, init=skip)
    float* alphab = aggbuf + (size_t)NN * 128;  // NE*2
    int*   mord   = (int*)(alphab + (size_t)NE * 2);  // NN*2
    float* den    = (float*)mord + (size_t)NN * 2;    // NN*2

    const int GEMM_BLK = 128;                    // 4 waves/block
    const int EDGE_BLK = 128;                    // 4 edges/block
    int edge_grid = NE / 4;                      // 800000 % 4 == 0

    auto gemm_grid = [](int nrows, int m) {
        int tiles = ((nrows + 15) / 16) * (m / 16);
        return (tiles + 3) / 4;
    };

    // ---------------- Layer 1 (H=2, C=64, D=128) ----------------
    wmma_gemm_bias<<<gemm_grid(NN, 128), GEMM_BLK, 0, stream>>>(x, Wq1, bq1, qbuf, NN, 128, 128, 0);
    wmma_gemm_bias<<<gemm_grid(NN, 128), GEMM_BLK, 0, stream>>>(x, Wk1, bk1, kbuf, NN, 128, 128, 0);
    wmma_gemm_bias<<<gemm_grid(NN, 128), GEMM_BLK, 0, stream>>>(x, Wv1, bv1, vbuf, NN, 128, 128, 0);
    wmma_gemm_bias<<<gemm_grid(NN, 128), GEMM_BLK, 0, stream>>>(x, Ws1, bs1, aggbuf, NN, 128, 128, 0);

    int nh1 = NN * 2;
    init_m_den<<<(nh1 + 255) / 256, 256, 0, stream>>>(mord, den, nh1);
    edge_alpha<2, 128><<<edge_grid, EDGE_BLK, 0, stream>>>(src, dst, eattr, We1, qbuf, kbuf,
                                                           alphab, mord, NE);
    finalize_m<<<(nh1 + 255) / 256, 256, 0, stream>>>(mord, nh1);
    edge_exp<<<(NE * 2 + 255) / 256, 256, 0, stream>>>(dst, alphab, (const float*)mord, den, NE, 2);
    edge_agg<2, 128><<<edge_grid, EDGE_BLK, 0, stream>>>(src, dst, eattr, We1, vbuf, alphab,
                                                         den, aggbuf, NE);
    relu_copy<<<((NN * 128) + 255) / 256, 256, 0, stream>>>(aggbuf, hbuf, NN * 128);

    // ---------------- Layer 2 (H=1, C=64, D=64) ----------------
    float* out = (float*)d_out;  // used directly as agg buffer (init = skip term)
    wmma_gemm_bias<<<gemm_grid(NN, 64), GEMM_BLK, 0, stream>>>(hbuf, Wq2, bq2, qbuf, NN, 128, 64, 0);
    wmma_gemm_bias<<<gemm_grid(NN, 64), GEMM_BLK, 0, stream>>>(hbuf, Wk2, bk2, kbuf, NN, 128, 64, 0);
    wmma_gemm_bias<<<gemm_grid(NN, 64), GEMM_BLK, 0, stream>>>(hbuf, Wv2, bv2, vbuf, NN, 128, 64, 0);
    wmma_gemm_bias<<<gemm_grid(NN, 64), GEMM_BLK, 0, stream>>>(hbuf, Ws2, bs2, out, NN, 128, 64, 0);

    int nh2 = NN;
    init_m_den<<<(nh2 + 255) / 256, 256, 0, stream>>>(mord, den, nh2);
    edge_alpha<1, 64><<<edge_grid, EDGE_BLK, 0, stream>>>(src, dst, eattr, We2, qbuf, kbuf,
                                                          alphab, mord, NE);
    finalize_m<<<(nh2 + 255) / 256, 256, 0, stream>>>(mord, nh2);
    edge_exp<<<(NE + 255) / 256, 256, 0, stream>>>(dst, alphab, (const float*)mord, den, NE, 1);
    edge_agg<1, 64><<<edge_grid, EDGE_BLK, 0, stream>>>(src, dst, eattr, We2, vbuf, alphab,
                                                        den, out, NE);
}